// SGC_31233002176551
// MI455X (gfx1250) — compile-verified
//
#include <hip/hip_runtime.h>

typedef __attribute__((ext_vector_type(2))) float v2f;
typedef __attribute__((ext_vector_type(8))) float v8f;

#define N_NODES 100000
#define N_EDGES 1250000
#define DIM 64

// ---------------------------------------------------------------------------
// Zero the aggregation scratch buffer.
// ---------------------------------------------------------------------------
__global__ __launch_bounds__(256) void zero_f32(float* __restrict__ p, int n) {
  int i = blockIdx.x * blockDim.x + threadIdx.x;
  if (i < n) p[i] = 0.0f;
}

// ---------------------------------------------------------------------------
// Scatter-add: agg[dst[e]] += h[src[e]], 16 threads per edge, float4 each.
// Edge indices are int64 per the reference. Atomics hit the L2-resident
// 25.6MB agg buffer (fits in 192MB L2).
// ---------------------------------------------------------------------------
__global__ __launch_bounds__(256) void scatter_add(
    const float* __restrict__ h, const long long* __restrict__ src,
    const long long* __restrict__ dst, float* __restrict__ agg) {
  int t = blockIdx.x * blockDim.x + threadIdx.x;
  int e = t >> 4;       // edge id
  int part = t & 15;    // which float4 of the 64-float row
  if (e >= N_EDGES) return;
  long long s = src[e];
  long long d = dst[e];
  const float4 v = ((const float4*)(h + (size_t)s * DIM))[part];
  float* op = agg + (size_t)d * DIM + part * 4;
  atomicAdd(op + 0, v.x);
  atomicAdd(op + 1, v.y);
  atomicAdd(op + 2, v.z);
  atomicAdd(op + 3, v.w);
}

// ---------------------------------------------------------------------------
// out[n, j] = sum_k agg[n, k] * W[j, k] + b[j]
// One wave per 16-node tile; 4 column tiles x 16 K-steps of
// V_WMMA_F32_16X16X4_F32. W (16KB) staged in LDS.
//
// A 16x4 f32 layout: lanes 0-15 hold rows M with VGPR0=K0,VGPR1=K1;
//                    lanes 16-31 hold VGPR0=K2,VGPR1=K3.   -> float2 load
// B 4x16 f32 layout: VGPR0: lanes0-15 K=0 / lanes16-31 K=2;
//                    VGPR1: lanes0-15 K=1 / lanes16-31 K=3. -> float2 from LDS
// C/D 16x16 f32:     VGPR v: lanes0-15 M=v, lanes16-31 M=v+8; N=lane%16.
// ---------------------------------------------------------------------------
__global__ __launch_bounds__(256) void ssg_gemm(
    const float* __restrict__ agg, const float* __restrict__ W,
    const float* __restrict__ b, float* __restrict__ out, int n_tiles) {
  __shared__ float sW[DIM * DIM];
  {
    const float4* s4 = (const float4*)W;
    float4* d4 = (float4*)sW;
    for (int i = threadIdx.x; i < (DIM * DIM) / 4; i += blockDim.x) d4[i] = s4[i];
  }
  __syncthreads();

  const int wave = threadIdx.x >> 5;
  const int lane = threadIdx.x & 31;
  const int tile = blockIdx.x * (blockDim.x >> 5) + wave;
  if (tile >= n_tiles) return;  // wave-uniform: EXEC stays all-ones for WMMA

  const int ln = lane & 15;            // row-in-tile (A) / column-in-tile (B,C)
  const int khalf = (lane >> 4) << 1;  // 0 for lanes 0-15, 2 for lanes 16-31

  v8f acc[4];
  acc[0] = (v8f){};
  acc[1] = (v8f){};
  acc[2] = (v8f){};
  acc[3] = (v8f){};

  const float* arow = agg + (size_t)(tile * 16 + ln) * DIM;

  for (int kb = 0; kb < 16; ++kb) {
    const int k0 = kb * 4 + khalf;
    v2f a = *(const v2f*)(arow + k0);  // A frag: K=k0, k0+1 for this lane half
    // B frags: B[k][j] = W[j][k]; lane covers column j = jt*16 + ln
    v2f b0 = *(const v2f*)(&sW[(0 * 16 + ln) * DIM + k0]);
    v2f b1 = *(const v2f*)(&sW[(1 * 16 + ln) * DIM + k0]);
    v2f b2 = *(const v2f*)(&sW[(2 * 16 + ln) * DIM + k0]);
    v2f b3 = *(const v2f*)(&sW[(3 * 16 + ln) * DIM + k0]);
    acc[0] = __builtin_amdgcn_wmma_f32_16x16x4_f32(false, a, false, b0,
                                                   (short)0, acc[0], false, false);
    acc[1] = __builtin_amdgcn_wmma_f32_16x16x4_f32(false, a, false, b1,
                                                   (short)0, acc[1], false, false);
    acc[2] = __builtin_amdgcn_wmma_f32_16x16x4_f32(false, a, false, b2,
                                                   (short)0, acc[2], false, false);
    acc[3] = __builtin_amdgcn_wmma_f32_16x16x4_f32(false, a, false, b3,
                                                   (short)0, acc[3], false, false);
  }

  // Bias + store. Lane covers column N = ln; VGPR v covers row M = v (+8 for
  // upper half-wave).
  const int mbase = tile * 16 + ((lane >> 4) << 3);
#pragma unroll
  for (int jt = 0; jt < 4; ++jt) {
    const float bias = b[jt * 16 + ln];
    float* op = out + (size_t)mbase * DIM + jt * 16 + ln;
#pragma unroll
    for (int v = 0; v < 8; ++v) {
      op[(size_t)v * DIM] = acc[jt][v] + bias;
    }
  }
}

// ---------------------------------------------------------------------------
// Orchestration: zero -> scatter -> gemm (hid), zero -> scatter -> gemm (out)
// d_out = [out (N*D) | hid (N*D)] per reference return order (out, hid).
// ---------------------------------------------------------------------------
extern "C" void kernel_launch(void* const* d_in, const int* in_sizes, int n_in,
                              void* d_out, int out_size, void* d_ws,
                              size_t ws_size, hipStream_t stream) {
  const float* x = (const float*)d_in[0];
  const long long* ei = (const long long*)d_in[1];
  const float* W1 = (const float*)d_in[2];
  const float* b1 = (const float*)d_in[3];
  const long long* src = ei;             // edge_index[0]
  const long long* dst = ei + N_EDGES;   // edge_index[1]

  float* out = (float*)d_out;
  float* hid = (float*)d_out + (size_t)N_NODES * DIM;
  float* agg = (float*)d_ws;  // 25.6 MB scratch, reused for both layers

  const int nelem = N_NODES * DIM;
  dim3 blk(256);
  dim3 zgrid((nelem + 255) / 256);
  dim3 sgrid((N_EDGES * 16 + 255) / 256);
  const int n_tiles = N_NODES / 16;  // 6250, exact
  dim3 ggrid((n_tiles + 7) / 8);     // 8 waves per block, 1 tile per wave

  // Layer 1: hid = (scatter_add(x)) @ W1^T + b1
  zero_f32<<<zgrid, blk, 0, stream>>>(agg, nelem);
  scatter_add<<<sgrid, blk, 0, stream>>>(x, src, dst, agg);
  ssg_gemm<<<ggrid, blk, 0, stream>>>(agg, W1, b1, hid, n_tiles);

  // Layer 2: out = (scatter_add(hid)) @ W1^T + b1
  zero_f32<<<zgrid, blk, 0, stream>>>(agg, nelem);
  scatter_add<<<sgrid, blk, 0, stream>>>(hid, src, dst, agg);
  ssg_gemm<<<ggrid, blk, 0, stream>>>(agg, W1, b1, out, n_tiles);
}